// GCN_412316860801
// MI455X (gfx1250) — compile-verified
//
#include <hip/hip_runtime.h>

// GCN (3x GCNConv) for MI455X / gfx1250, wave32.
// Inputs (setup_inputs order):
//  0: x [50000,256] f32     1: edge_index [2,800000] i32   2: edge_weight [800000] f32
//  3: W1 [256,128]          4: b1 [128]
//  5: W2 [128,128]          6: b2 [128]
//  7: W3 [128,128]          8: b3 [128]
// Output: [50000,128] f32

#define NNODES 50000
#define NEDGES 800000
#define INCH   256
#define HIDCH  128   // all layer outputs are 128 channels

typedef float v2f __attribute__((ext_vector_type(2)));
typedef float v8f __attribute__((ext_vector_type(8)));

// ---------------------------------------------------------------------------
// deg[i] = 1.0 (self loop weight), then atomic += ew over dst, then rsqrt.
// ---------------------------------------------------------------------------
__global__ __launch_bounds__(256) void k_init_deg(float* __restrict__ deg, int n) {
  int i = blockIdx.x * blockDim.x + threadIdx.x;
  if (i < n) deg[i] = 1.0f;
}

__global__ __launch_bounds__(256) void k_deg_accum(const int* __restrict__ dst,
                                                   const float* __restrict__ ew,
                                                   float* __restrict__ deg, int e) {
  int i = blockIdx.x * blockDim.x + threadIdx.x;
  if (i < e) unsafeAtomicAdd(deg + dst[i], ew[i]);   // hw global_atomic_add_f32
}

__global__ __launch_bounds__(256) void k_rsqrt(float* __restrict__ deg, int n) {
  int i = blockIdx.x * blockDim.x + threadIdx.x;
  if (i < n) deg[i] = rsqrtf(deg[i]);                // deg >= 1 always
}

__global__ __launch_bounds__(256) void k_zero4(float4* __restrict__ p, long long n4) {
  long long i = (long long)blockIdx.x * blockDim.x + threadIdx.x;
  if (i < n4) p[i] = make_float4(0.f, 0.f, 0.f, 0.f);
}

// ---------------------------------------------------------------------------
// C[M,N] = A[M,K] @ B[K,N], fp32 V_WMMA_F32_16X16X4_F32.
// One wave computes a 16(M) x 128(N) strip = 8 accumulator tiles, so the A
// fragment is loaded once per k-step and amortized over 8 WMMAs
// (2 + 16 loads per 8 WMMAs vs 4 loads/WMMA for tile-per-wave).
// Fragment layouts (CDNA5 ISA 7.12.2):
//   A 16x4:  lane = m + 16*(k>>1), vgpr = k&1
//   B 4x16:  lane = n + 16*(k>>1), vgpr = k&1
//   C 16x16: lane = n + 16*(m>=8), vgpr r -> m = r + 8*(lane>>4)
// M multiple of 16 (50000 = 3125*16), N = 128, K in {256,128}.
// ---------------------------------------------------------------------------
__global__ __launch_bounds__(128) void k_gemm_wmma_f32(const float* __restrict__ A,
                                                       const float* __restrict__ B,
                                                       float* __restrict__ C,
                                                       int M, int N, int K) {
  const int lane  = threadIdx.x & 31;
  const int wave  = threadIdx.x >> 5;
  const int strip = blockIdx.x * 4 + wave;           // 16-row strip index
  if (strip >= (M >> 4)) return;                     // whole-wave exit; EXEC all-1s

  const int mn   = lane & 15;                        // row (A) / col (B,C) in tile
  const int kSel = (lane >> 4) * 2;                  // lanes 16..31 hold K+2,K+3

  const float* __restrict__ arow  = A + (size_t)(strip * 16 + mn) * K;
  const float* __restrict__ bbase = B + mn;

  v8f acc[8];
#pragma unroll
  for (int t = 0; t < 8; ++t) acc[t] = (v8f){};

  for (int k0 = 0; k0 < K; k0 += 4) {
    const int ka = k0 + kSel;
    v2f a;
    a.x = arow[ka];
    a.y = arow[ka + 1];
    const float* __restrict__ b0 = bbase + (size_t)ka * N;
    const float* __restrict__ b1 = bbase + (size_t)(ka + 1) * N;
#pragma unroll
    for (int t = 0; t < 8; ++t) {
      v2f b;
      b.x = b0[t * 16];
      b.y = b1[t * 16];
      acc[t] = __builtin_amdgcn_wmma_f32_16x16x4_f32(
          /*neg_a=*/false, a, /*neg_b=*/false, b,
          /*c_mod=*/(short)0, acc[t], /*reuse_a=*/false, /*reuse_b=*/false);
    }
  }

  const int mbase = (lane >> 4) * 8;
  float* __restrict__ cbase = C + (size_t)(strip * 16) * N + mn;
#pragma unroll
  for (int t = 0; t < 8; ++t) {
    float* __restrict__ crow = cbase + t * 16;
#pragma unroll
    for (int r = 0; r < 8; ++r)
      crow[(size_t)(mbase + r) * N] = acc[t][r];
  }
}

// ---------------------------------------------------------------------------
// Edge scatter: one wave per edge, 4 channels per lane (128 ch).
// agg[dst, c] += dinv[src]*ew*dinv[dst] * h[src, c]
// h (25.6 MB) is L2-resident on the 192 MB L2 -> L2-local gathers + atomics.
// ---------------------------------------------------------------------------
__global__ __launch_bounds__(256) void k_scatter(const int* __restrict__ src,
                                                 const int* __restrict__ dst,
                                                 const float* __restrict__ ew,
                                                 const float* __restrict__ dinv,
                                                 const float* __restrict__ h,
                                                 float* __restrict__ agg, int E) {
  long long gid = (long long)blockIdx.x * blockDim.x + threadIdx.x;
  int e = (int)(gid >> 5);
  if (e >= E) return;
  int q = (int)(gid & 31);                            // 4-channel chunk
  int s = src[e];
  int d = dst[e];
  float norm = dinv[s] * ew[e] * dinv[d];
  const float4 hv = *(const float4*)(h + (size_t)s * HIDCH + q * 4);
  float* out = agg + (size_t)d * HIDCH + q * 4;
  unsafeAtomicAdd(out + 0, norm * hv.x);
  unsafeAtomicAdd(out + 1, norm * hv.y);
  unsafeAtomicAdd(out + 2, norm * hv.z);
  unsafeAtomicAdd(out + 3, norm * hv.w);
}

// ---------------------------------------------------------------------------
// out[i,c] = agg[i,c] + dinv[i]^2 * h[i,c] + b[c]   (+ ReLU except last layer)
// ---------------------------------------------------------------------------
__global__ __launch_bounds__(256) void k_finalize(const float* __restrict__ agg,
                                                  const float* __restrict__ h,
                                                  const float* __restrict__ dinv,
                                                  const float* __restrict__ bias,
                                                  float* __restrict__ out,
                                                  long long total, int doRelu) {
  long long idx = (long long)blockIdx.x * blockDim.x + threadIdx.x;
  if (idx >= total) return;
  int i = (int)(idx >> 7);                            // C = 128
  int c = (int)(idx & 127);
  float di = dinv[i];
  float v = agg[idx] + di * di * h[idx] + bias[c];
  out[idx] = doRelu ? fmaxf(v, 0.f) : v;
}

// ---------------------------------------------------------------------------
extern "C" void kernel_launch(void* const* d_in, const int* in_sizes, int n_in,
                              void* d_out, int out_size, void* d_ws, size_t ws_size,
                              hipStream_t stream) {
  (void)in_sizes; (void)n_in; (void)out_size; (void)ws_size;

  const float* x  = (const float*)d_in[0];
  const int*   ei = (const int*)d_in[1];
  const float* ew = (const float*)d_in[2];
  const float* W1 = (const float*)d_in[3];
  const float* b1 = (const float*)d_in[4];
  const float* W2 = (const float*)d_in[5];
  const float* b2 = (const float*)d_in[6];
  const float* W3 = (const float*)d_in[7];
  const float* b3 = (const float*)d_in[8];
  float* out = (float*)d_out;

  const int* src = ei;            // edge_index[0, :]
  const int* dst = ei + NEDGES;   // edge_index[1, :]

  // workspace layout (floats)
  float* ws   = (float*)d_ws;
  float* dinv = ws;                                       // 50000
  float* h    = ws + 50048;                               // 6.4M (aligned start)
  float* agg  = h + (size_t)NNODES * HIDCH;               // 6.4M
  float* xb   = agg + (size_t)NNODES * HIDCH;             // 6.4M ping buffer

  // --- symmetric normalization: dinv = rsqrt(1 + segsum(ew, dst)) ---
  k_init_deg<<<(NNODES + 255) / 256, 256, 0, stream>>>(dinv, NNODES);
  k_deg_accum<<<(NEDGES + 255) / 256, 256, 0, stream>>>(dst, ew, dinv, NEDGES);
  k_rsqrt<<<(NNODES + 255) / 256, 256, 0, stream>>>(dinv, NNODES);

  const int strips = NNODES / 16;                         // 3125 row strips
  const int gemmBlocks = (strips + 3) / 4;                // 4 waves/block
  const long long aggN4 = (long long)NNODES * HIDCH / 4;
  const long long scatThreads = (long long)NEDGES * 32;
  const long long total = (long long)NNODES * HIDCH;

  auto layer = [&](const float* Ain, int K, const float* W, const float* bias,
                   float* outp, int relu) {
    k_gemm_wmma_f32<<<gemmBlocks, 128, 0, stream>>>(Ain, W, h, NNODES, HIDCH, K);
    k_zero4<<<(int)((aggN4 + 255) / 256), 256, 0, stream>>>((float4*)agg, aggN4);
    k_scatter<<<(int)((scatThreads + 255) / 256), 256, 0, stream>>>(
        src, dst, ew, dinv, h, agg, NEDGES);
    k_finalize<<<(int)((total + 255) / 256), 256, 0, stream>>>(
        agg, h, dinv, bias, outp, total, relu);
  };

  layer(x,  INCH,  W1, b1, xb,  1);   // conv1 + relu
  layer(xb, HIDCH, W2, b2, xb,  1);   // conv2 + relu (in-place ping OK: gemm reads xb first)
  layer(xb, HIDCH, W3, b3, out, 0);   // conv3 -> d_out
}